// SwitchGate_16819091931299
// MI455X (gfx1250) — compile-verified
//
#include <hip/hip_runtime.h>
#include <math.h>

// Problem constants (static in the reference)
#define S_TOK 4096
#define DDIM  2048
#define NEXP  64      // NUM_EXPERTS (router outputs)
#define PROT  4       // prototypes
#define EXP   16      // experts per prototype = NEXP / PROT
#define CAP   320     // ceil(1.25 * S / E)

typedef float v2f __attribute__((ext_vector_type(2)));
typedef float v4f __attribute__((ext_vector_type(4)));
typedef float v8f __attribute__((ext_vector_type(8)));

// ---------------------------------------------------------------------------
// 1) Streaming zero-fill of the entire output (671 MB => the real bottleneck,
//    ~29us at 23.3 TB/s). Nontemporal 16B stores; no reuse wanted.
// ---------------------------------------------------------------------------
__global__ void zero_out_kernel(float* __restrict__ out, long long n) {
    long long i4 = ((long long)blockIdx.x * blockDim.x + threadIdx.x) * 4;
    if (i4 + 3 < n) {
        v4f z = {0.f, 0.f, 0.f, 0.f};
        __builtin_nontemporal_store(z, (v4f*)(out + i4));
    } else if (i4 < n) {
        for (long long j = i4; j < n; ++j) out[j] = 0.f;
    }
}

__global__ void init_ws_kernel(float* __restrict__ me_sum) {
    me_sum[threadIdx.x] = 0.f;   // 64 accumulators
}

// ---------------------------------------------------------------------------
// 2) Router GEMM: logits[s][n] = sum_k x[s][k] * w[n][k]  (fp32, exact match
//    to reference precision) via V_WMMA_F32_16X16X4_F32.
//    Block = 128 threads (4 waves). Wave w owns the 16-col tile n0 = 16*w,
//    block b owns the 16-row tile m0 = 16*b. K-loop: 2048/4 = 512 WMMAs.
//    Fragment layouts per CDNA5 ISA 7.12.2 (32-bit A 16x4 / B 4x16 / C 16x16).
// ---------------------------------------------------------------------------
__global__ void gemm_wmma_f32_kernel(const float* __restrict__ X,
                                     const float* __restrict__ W,
                                     float* __restrict__ logits) {
    const int tid  = threadIdx.x;
    const int lane = tid & 31;
    const int wave = tid >> 5;
    const int m0   = blockIdx.x * 16;
    const int n0   = wave * 16;
    const int r16  = lane & 15;       // M (for A) / N (for B) within tile
    const int kb   = (lane >> 4) * 2; // K sub-offset {0,2}

    const float* xp = X + (size_t)(m0 + r16) * DDIM + kb; // A: row m, K=k+kb..+1
    const float* wp = W + (size_t)(n0 + r16) * DDIM + kb; // B: col n (W is [N,D])

    v8f c = {};
    for (int k = 0; k < DDIM; k += 4) {
        v2f a = *(const v2f*)(xp + k);   // 8B-aligned (k+kb even)
        v2f b = *(const v2f*)(wp + k);
        c = __builtin_amdgcn_wmma_f32_16x16x4_f32(
            /*neg_a=*/false, a, /*neg_b=*/false, b,
            /*c_mod=*/(short)0, c, /*reuse_a=*/false, /*reuse_b=*/false);
    }

    // C/D layout: VGPR r -> M = r + 8*(lane>>4), N = lane&15
    const int col   = n0 + (lane & 15);
    const int rbase = 8 * (lane >> 4);
    for (int r = 0; r < 8; ++r)
        logits[(size_t)(m0 + r + rbase) * NEXP + col] = c[r];
}

// ---------------------------------------------------------------------------
// 3) Fused softmax (fp32) + argmax per (s,p); LDS partial sums of gates for
//    the l_aux "me" term (ds_add_f32), flushed with 64 global atomics/block.
// ---------------------------------------------------------------------------
__global__ void softmax_argmax_kernel(const float* __restrict__ logits,
                                      float* __restrict__ gates,
                                      int* __restrict__ idx,
                                      float* __restrict__ me_sum) {
    __shared__ float sm[PROT * EXP];
    const int t = threadIdx.x;
    if (t < PROT * EXP) sm[t] = 0.f;
    __syncthreads();

    const int gid = blockIdx.x * blockDim.x + t;   // [0, S*P)
    const int s = gid / PROT, p = gid % PROT;
    const float* lp = logits + (size_t)s * NEXP + p * EXP;

    float v[EXP];
    float mx = -INFINITY;
    for (int e = 0; e < EXP; ++e) { v[e] = lp[e]; mx = fmaxf(mx, v[e]); }
    float sum = 0.f;
    for (int e = 0; e < EXP; ++e) { v[e] = expf(v[e] - mx); sum += v[e]; }
    const float inv = 1.f / sum;

    float* gp = gates + (size_t)s * NEXP + p * EXP;
    int best = 0; float bv = -INFINITY;
    for (int e = 0; e < EXP; ++e) {
        const float g = v[e] * inv;
        gp[e] = g;
        atomicAdd(&sm[p * EXP + e], g);            // LDS ds_add_f32
        if (g > bv) { bv = g; best = e; }          // first-max wins (ties)
    }
    idx[gid] = best;

    __syncthreads();
    if (t < PROT * EXP) atomicAdd(&me_sum[t], sm[t]);
}

// ---------------------------------------------------------------------------
// 4) Token-order capacity scan: one wave32 per (p,e) column; 4096 tokens in
//    128 ballot+popc steps. Writes each kept token's slot rank (or -1) and
//    the pre-drop selection count (for "ce" in l_aux).
// ---------------------------------------------------------------------------
__global__ void capacity_scan_kernel(const int* __restrict__ idx,
                                     int* __restrict__ loc,
                                     int* __restrict__ cnt) {
    const int lane = threadIdx.x;               // blockDim.x == 32
    const int p = blockIdx.x / EXP, e = blockIdx.x % EXP;
    const unsigned lt = (1u << lane) - 1u;
    int base = 0;
    for (int chunk = 0; chunk < S_TOK / 32; ++chunk) {
        const int s = chunk * 32 + lane;
        const bool match = (idx[s * PROT + p] == e);
        const unsigned m = __builtin_amdgcn_ballot_w32(match);
        if (match) {
            const int rank = base + __popc(m & lt);
            loc[s * PROT + p] = (rank < CAP) ? rank : -1;
        }
        base += __popc(m);
    }
    if (lane == 0) cnt[blockIdx.x] = base;
}

// ---------------------------------------------------------------------------
// 5) Sparse scatter: at most one nonzero per (s,p) into combine_weights and
//    dispatch_mask (<=32K stores vs 168M zeros).
// ---------------------------------------------------------------------------
__global__ void scatter_kernel(const float* __restrict__ gates,
                               const int* __restrict__ idx,
                               const int* __restrict__ loc,
                               float* __restrict__ out) {
    const int gid = blockIdx.x * blockDim.x + threadIdx.x;
    if (gid >= S_TOK * PROT) return;
    const int s = gid / PROT, p = gid % PROT;
    const int e = idx[gid];
    const int c = loc[gid];
    if (c < 0) return;                           // over-capacity: row stays 0
    const float g = gates[(size_t)s * NEXP + p * EXP + e];
    const size_t off = ((size_t)gid * EXP + e) * CAP + c;
    float* combine  = out + 1;
    float* dispatch = out + 1 + (size_t)S_TOK * PROT * EXP * CAP;
    combine[off]  = g;
    dispatch[off] = (g != 0.f) ? 1.f : 0.f;
}

// ---------------------------------------------------------------------------
// 6) l_aux = mean(me * ce) over the 64 (p,e) cells.
// ---------------------------------------------------------------------------
__global__ void laux_kernel(const float* __restrict__ me_sum,
                            const int* __restrict__ cnt,
                            float* __restrict__ out) {
    __shared__ float sm[PROT * EXP];
    const int t = threadIdx.x;                   // blockDim.x == 64
    const float me = me_sum[t] * (1.f / (float)S_TOK);
    const float ce = (float)cnt[t] * (1.f / (float)S_TOK);
    sm[t] = me * ce;
    __syncthreads();
    for (int o = 32; o > 0; o >>= 1) {
        if (t < o) sm[t] += sm[t + o];
        __syncthreads();
    }
    if (t == 0) out[0] = sm[0] * (1.f / (float)(PROT * EXP));
}

// ---------------------------------------------------------------------------
extern "C" void kernel_launch(void* const* d_in, const int* in_sizes, int n_in,
                              void* d_out, int out_size, void* d_ws, size_t ws_size,
                              hipStream_t stream) {
    const float* x = (const float*)d_in[0];      // [S, D]
    const float* w = (const float*)d_in[1];      // [N, D]
    float* out = (float*)d_out;

    // workspace layout (~1.2 MB)
    char* ws = (char*)d_ws;
    float* gates  = (float*)ws;                                   // S*64 f32 (logits reused in-place)
    int*   idx    = (int*)(ws + (size_t)S_TOK * NEXP * sizeof(float));
    int*   loc    = idx + S_TOK * PROT;
    float* me_sum = (float*)(loc + S_TOK * PROT);
    int*   cnt    = (int*)(me_sum + PROT * EXP);

    const long long ntotal = 1LL + 2LL * S_TOK * PROT * EXP * CAP; // == out_size
    const long long n4 = (ntotal + 3) / 4;

    zero_out_kernel<<<(unsigned)((n4 + 255) / 256), 256, 0, stream>>>(out, ntotal);
    init_ws_kernel<<<1, PROT * EXP, 0, stream>>>(me_sum);
    gemm_wmma_f32_kernel<<<S_TOK / 16, 128, 0, stream>>>(x, w, gates);
    softmax_argmax_kernel<<<(S_TOK * PROT) / 256, 256, 0, stream>>>(gates, gates, idx, me_sum);
    capacity_scan_kernel<<<PROT * EXP, 32, 0, stream>>>(idx, loc, cnt);
    scatter_kernel<<<(S_TOK * PROT + 255) / 256, 256, 0, stream>>>(gates, idx, loc, out);
    laux_kernel<<<1, PROT * EXP, 0, stream>>>(me_sum, cnt, out);
}